// SAM3GuidedCLDiceLoss_35862976921962
// MI455X (gfx1250) — compile-verified
//
#include <hip/hip_runtime.h>
#include <math.h>
#include <stdint.h>

#define H_IMG 512
#define W_IMG 512
#define NB    8
#define NC    19
#define HW    (H_IMG * W_IMG)          // 262144
#define NPIX  ((size_t)NB * HW)        // 2097152
#define ITERS 25
#define EPS_F 1e-6f

#define TS  32
#define RAD 3
#define IW  (TS + 2 * RAD)             // 38
#define LST 40                          // LDS row stride (pad vs. bank conflicts)

// ---- CDNA5 async global->LDS staging (ASYNCcnt path) ----------------------
#if defined(__gfx1250__) && __has_builtin(__builtin_amdgcn_global_load_async_to_lds_b32)
#define HAVE_ASYNC 1
#else
#define HAVE_ASYNC 0
#endif

__device__ __forceinline__ void async_copy_f32(const float* g, float* l) {
#if HAVE_ASYNC
  __builtin_amdgcn_global_load_async_to_lds_b32(
      (__attribute__((address_space(1))) int*)(uintptr_t)g,
      (__attribute__((address_space(3))) int*)(uintptr_t)l,
      0, 0);
#else
  *l = *g;
#endif
}

__device__ __forceinline__ void async_wait_all() {
#if HAVE_ASYNC
#if __has_builtin(__builtin_amdgcn_s_wait_asynccnt)
  __builtin_amdgcn_s_wait_asynccnt(0);
#else
  asm volatile("s_wait_asynccnt 0" ::: "memory");
#endif
#endif
}

// ---- 3x3 LDS min / max ----------------------------------------------------
__device__ __forceinline__ float min9(const float* L, int r, int c) {
  const float* p0 = L + (r - 1) * LST + (c - 1);
  const float* p1 = p0 + LST;
  const float* p2 = p1 + LST;
  float m = fminf(fminf(p0[0], p0[1]), p0[2]);
  m = fminf(m, fminf(fminf(p1[0], p1[1]), p1[2]));
  return fminf(m, fminf(fminf(p2[0], p2[1]), p2[2]));
}

__device__ __forceinline__ float max9(const float* L, int r, int c) {
  const float* p0 = L + (r - 1) * LST + (c - 1);
  const float* p1 = p0 + LST;
  const float* p2 = p1 + LST;
  float m = fmaxf(fmaxf(p0[0], p0[1]), p0[2]);
  m = fmaxf(m, fmaxf(fmaxf(p1[0], p1[1]), p1[2]));
  return fmaxf(m, fmaxf(fmaxf(p2[0], p2[1]), p2[2]));
}

// ---- fused morphology iteration (the hot kernel; placed first in the TU) --
// im' = erode(im); delta = relu(im' - dilate(erode(im')));
// skel += relu(delta - skel*delta); write im'
// SAME padding: erode pads +inf, dilate pads -inf -> out-of-image halo cells
// are filled with the identity of the NEXT reduction stage.
__global__ __launch_bounds__(256)
void morph_iter(const float* __restrict__ im_s, const float* __restrict__ im_t,
                float* __restrict__ imo_s, float* __restrict__ imo_t,
                float* __restrict__ sk_s,  float* __restrict__ sk_t) {
  __shared__ float A [IW * LST];
  __shared__ float Bm[IW * LST];
  const int tid = threadIdx.x;
  const int z   = blockIdx.z;            // 0..15 : batch x {student,teacher}
  const int b   = z & 7;
  const bool isT = z >= 8;
  const float* im = (isT ? im_t : im_s) + (size_t)b * HW;
  float* imo      = (isT ? imo_t : imo_s) + (size_t)b * HW;
  float* sk       = (isT ? sk_t : sk_s) + (size_t)b * HW;
  const int x0 = blockIdx.x * TS, y0 = blockIdx.y * TS;
  const float PINF = __builtin_inff();
  const bool edge = (x0 == 0) || (y0 == 0) || (x0 + TS == W_IMG) || (y0 + TS == H_IMG);

  // stage 0: stage im tile (+halo 3) into LDS via async global->LDS
  if (!edge) {
    for (int i = tid; i < IW * IW; i += 256) {
      int r = i / IW, c = i - (i / IW) * IW;
      async_copy_f32(&im[(y0 + r - RAD) * W_IMG + (x0 + c - RAD)], &A[r * LST + c]);
    }
  } else {
    for (int i = tid; i < IW * IW; i += 256) {
      int r = i / IW, c = i - (i / IW) * IW;
      int gy = y0 + r - RAD, gx = x0 + c - RAD;
      float* lp = &A[r * LST + c];
      if ((unsigned)gy < (unsigned)H_IMG && (unsigned)gx < (unsigned)W_IMG)
        async_copy_f32(&im[gy * W_IMG + gx], lp);
      else
        *lp = PINF;                      // identity for first erode (min)
    }
  }
  async_wait_all();
  __syncthreads();

  // stage 1: e1 = erode(A) on [1, IW-1)^2   (OOB -> +inf, next stage is min)
  if (!edge) {
    for (int i = tid; i < 36 * 36; i += 256) {
      int r0 = i / 36, c0 = i - r0 * 36;
      Bm[(r0 + 1) * LST + (c0 + 1)] = min9(A, r0 + 1, c0 + 1);
    }
  } else {
    for (int i = tid; i < 36 * 36; i += 256) {
      int r0 = i / 36, c0 = i - r0 * 36;
      int r = r0 + 1, c = c0 + 1;
      int gy = y0 + r - RAD, gx = x0 + c - RAD;
      float v = ((unsigned)gy < (unsigned)H_IMG && (unsigned)gx < (unsigned)W_IMG)
                    ? min9(A, r, c) : PINF;
      Bm[r * LST + c] = v;
    }
  }
  __syncthreads();

  // stage 2: e2 = erode(e1) on [2, IW-2)^2 -> reuse A  (OOB -> -inf, next is max)
  if (!edge) {
    for (int i = tid; i < 34 * 34; i += 256) {
      int r0 = i / 34, c0 = i - r0 * 34;
      A[(r0 + 2) * LST + (c0 + 2)] = min9(Bm, r0 + 2, c0 + 2);
    }
  } else {
    for (int i = tid; i < 34 * 34; i += 256) {
      int r0 = i / 34, c0 = i - r0 * 34;
      int r = r0 + 2, c = c0 + 2;
      int gy = y0 + r - RAD, gx = x0 + c - RAD;
      float v = ((unsigned)gy < (unsigned)H_IMG && (unsigned)gx < (unsigned)W_IMG)
                    ? min9(Bm, r, c) : -PINF;
      A[r * LST + c] = v;
    }
  }
  __syncthreads();

  // stage 3: d = dilate(e2); delta = relu(e1 - d); skel RMW; write im' = e1
  // (interior tile positions are always in-image: 512 % 32 == 0)
#pragma unroll
  for (int k = 0; k < 4; ++k) {
    int i = tid + k * 256;
    int r = i >> 5, c = i & 31;
    int lr = r + RAD, lc = c + RAD;
    float d  = max9(A, lr, lc);
    float e1 = Bm[lr * LST + lc];
    float delta = fmaxf(e1 - d, 0.f);
    size_t gi = (size_t)(y0 + r) * W_IMG + (x0 + c);
    float s = sk[gi];
    sk[gi]  = s + fmaxf(delta - s * delta, 0.f);
    imo[gi] = e1;
  }
}

// ---- initial skeleton: skel = relu(im - dilate(erode(im))) ----------------
__global__ __launch_bounds__(256)
void morph_init(const float* __restrict__ im_s, const float* __restrict__ im_t,
                float* __restrict__ sk_s,  float* __restrict__ sk_t) {
  __shared__ float A [IW * LST];
  __shared__ float Bm[IW * LST];
  const int tid = threadIdx.x;
  const int z   = blockIdx.z;
  const int b   = z & 7;
  const bool isT = z >= 8;
  const float* im = (isT ? im_t : im_s) + (size_t)b * HW;
  float* sk       = (isT ? sk_t : sk_s) + (size_t)b * HW;
  const int x0 = blockIdx.x * TS, y0 = blockIdx.y * TS;
  const float PINF = __builtin_inff();
  const bool edge = (x0 == 0) || (y0 == 0) || (x0 + TS == W_IMG) || (y0 + TS == H_IMG);

  if (!edge) {
    for (int i = tid; i < IW * IW; i += 256) {
      int r = i / IW, c = i - (i / IW) * IW;
      async_copy_f32(&im[(y0 + r - RAD) * W_IMG + (x0 + c - RAD)], &A[r * LST + c]);
    }
  } else {
    for (int i = tid; i < IW * IW; i += 256) {
      int r = i / IW, c = i - (i / IW) * IW;
      int gy = y0 + r - RAD, gx = x0 + c - RAD;
      float* lp = &A[r * LST + c];
      if ((unsigned)gy < (unsigned)H_IMG && (unsigned)gx < (unsigned)W_IMG)
        async_copy_f32(&im[gy * W_IMG + gx], lp);
      else
        *lp = PINF;
    }
  }
  async_wait_all();
  __syncthreads();

  // e1 = erode(A) on [1, IW-1)^2   (OOB -> -inf, next stage is max)
  if (!edge) {
    for (int i = tid; i < 36 * 36; i += 256) {
      int r0 = i / 36, c0 = i - r0 * 36;
      Bm[(r0 + 1) * LST + (c0 + 1)] = min9(A, r0 + 1, c0 + 1);
    }
  } else {
    for (int i = tid; i < 36 * 36; i += 256) {
      int r0 = i / 36, c0 = i - r0 * 36;
      int r = r0 + 1, c = c0 + 1;
      int gy = y0 + r - RAD, gx = x0 + c - RAD;
      float v = ((unsigned)gy < (unsigned)H_IMG && (unsigned)gx < (unsigned)W_IMG)
                    ? min9(A, r, c) : -PINF;
      Bm[r * LST + c] = v;
    }
  }
  __syncthreads();

  // d = dilate(e1); skel = relu(im - d)
#pragma unroll
  for (int k = 0; k < 4; ++k) {
    int i = tid + k * 256;
    int r = i >> 5, c = i & 31;
    int lr = r + RAD, lc = c + RAD;
    float d   = max9(Bm, lr, lc);
    float imv = A[lr * LST + lc];
    size_t gi = (size_t)(y0 + r) * W_IMG + (x0 + c);
    sk[gi] = fmaxf(imv - d, 0.f);
  }
}

// ---- prep: softmax class plane + teacher clip -----------------------------
__global__ __launch_bounds__(256)
void prep_student(const float* __restrict__ logits, const int* __restrict__ cidx,
                  float* __restrict__ prob, float* __restrict__ im0) {
  size_t p = (size_t)blockIdx.x * blockDim.x + threadIdx.x;
  if (p >= NPIX) return;
  int ci = *cidx;
  size_t b  = p / HW;
  size_t px = p - b * HW;
  const float* base = logits + b * (size_t)NC * HW + px;
  float v[NC];
  float m = -__builtin_inff();
#pragma unroll
  for (int c = 0; c < NC; ++c) {
    v[c] = base[(size_t)c * HW];
    m = fmaxf(m, v[c]);
  }
  float s = 0.f;
#pragma unroll
  for (int c = 0; c < NC; ++c) s += __expf(v[c] - m);
  float xc = base[(size_t)ci * HW];       // re-load selected channel (cache hit),
  float pr = __expf(xc - m) / s;          // avoids dynamic register indexing
  pr = fminf(fmaxf(pr, 0.f), 1.f);
  prob[p] = pr;
  im0[p]  = pr;
}

__global__ __launch_bounds__(256)
void prep_teacher(const float* __restrict__ mask,
                  float* __restrict__ prob, float* __restrict__ im0) {
  size_t p = (size_t)blockIdx.x * blockDim.x + threadIdx.x;
  if (p >= NPIX) return;
  float v = fminf(fmaxf(mask[p], 0.f), 1.f);
  prob[p] = v;
  im0[p]  = v;
}

// ---- reductions -----------------------------------------------------------
__global__ __launch_bounds__(256)
void reduce_sums(const float* __restrict__ s_skel, const float* __restrict__ t_prob,
                 const float* __restrict__ t_skel, const float* __restrict__ s_prob,
                 float* __restrict__ partial) {   // [8][16][4]
  const int b = blockIdx.y, j = blockIdx.x;       // j in 0..15
  const int CH = HW / 16;
  size_t base = (size_t)b * HW + (size_t)j * CH;
  float a0 = 0.f, a1 = 0.f, a2 = 0.f, a3 = 0.f;
  for (int i = threadIdx.x; i < CH; i += 256) {
    size_t idx = base + i;
    float ss = s_skel[idx], tp = t_prob[idx];
    float ts = t_skel[idx], sp = s_prob[idx];
    a0 = fmaf(ss, tp, a0); a1 += ss;
    a2 = fmaf(ts, sp, a2); a3 += ts;
  }
  __shared__ float red[4 * 256];
  int tid = threadIdx.x;
  red[tid] = a0; red[256 + tid] = a1; red[512 + tid] = a2; red[768 + tid] = a3;
  __syncthreads();
  for (int s = 128; s > 0; s >>= 1) {
    if (tid < s) {
      red[tid]       += red[tid + s];
      red[256 + tid] += red[256 + tid + s];
      red[512 + tid] += red[512 + tid + s];
      red[768 + tid] += red[768 + tid + s];
    }
    __syncthreads();
  }
  if (tid == 0) {
    float* o = partial + ((size_t)b * 16 + j) * 4;
    o[0] = red[0]; o[1] = red[256]; o[2] = red[512]; o[3] = red[768];
  }
}

__global__ void final_loss(const float* __restrict__ partial, float* __restrict__ out) {
  if (threadIdx.x != 0 || blockIdx.x != 0) return;
  float acc = 0.f;
  for (int b = 0; b < NB; ++b) {
    float A = 0.f, Bv = 0.f, C = 0.f, D = 0.f;
    for (int j = 0; j < 16; ++j) {
      const float* o = partial + ((size_t)b * 16 + j) * 4;
      A += o[0]; Bv += o[1]; C += o[2]; D += o[3];
    }
    float tprec = (A + EPS_F) / (Bv + EPS_F);
    float tsens = (C + EPS_F) / (D + EPS_F);
    float cl = (2.f * tprec * tsens + EPS_F) / (tprec + tsens + EPS_F);
    acc += 1.f - cl;
  }
  out[0] = acc * (1.f / NB);
}

// ---- launch ---------------------------------------------------------------
extern "C" void kernel_launch(void* const* d_in, const int* in_sizes, int n_in,
                              void* d_out, int out_size, void* d_ws, size_t ws_size,
                              hipStream_t stream) {
  const float* logits = (const float*)d_in[0];
  const float* mask   = (const float*)d_in[1];
  const int*   cidx   = (const int*)d_in[2];
  (void)in_sizes; (void)n_in; (void)out_size; (void)ws_size;

  float* w = (float*)d_ws;
  const size_t N = NPIX;
  float* s_prob  = w + 0 * N;
  float* t_prob  = w + 1 * N;
  float* s_imA   = w + 2 * N;
  float* s_imB   = w + 3 * N;
  float* t_imA   = w + 4 * N;
  float* t_imB   = w + 5 * N;
  float* s_sk    = w + 6 * N;
  float* t_sk    = w + 7 * N;
  float* partial = w + 8 * N;                      // 512 floats

  dim3 blk(256);
  int nblk = (int)((N + 255) / 256);               // 8192
  prep_student<<<nblk, blk, 0, stream>>>(logits, cidx, s_prob, s_imA);
  prep_teacher<<<nblk, blk, 0, stream>>>(mask, t_prob, t_imA);

  dim3 mgrid(W_IMG / TS, H_IMG / TS, 16);          // 16x16 tiles x (8 batch * 2 maps)
  morph_init<<<mgrid, blk, 0, stream>>>(s_imA, t_imA, s_sk, t_sk);
  for (int it = 0; it < ITERS - 1; ++it) {         // 24 fused iterations, ping-pong im
    float* ss = (it & 1) ? s_imB : s_imA;
    float* sd = (it & 1) ? s_imA : s_imB;
    float* ts = (it & 1) ? t_imB : t_imA;
    float* td = (it & 1) ? t_imA : t_imB;
    morph_iter<<<mgrid, blk, 0, stream>>>(ss, ts, sd, td, s_sk, t_sk);
  }

  reduce_sums<<<dim3(16, NB), blk, 0, stream>>>(s_sk, t_prob, t_sk, s_prob, partial);
  final_loss<<<1, 32, 0, stream>>>(partial, (float*)d_out);
}